// GATBackbone_23158463660629
// MI455X (gfx1250) — compile-verified
//
#include <hip/hip_runtime.h>
#include <hip/hip_bf16.h>

// Problem constants (fixed by the reference)
#define NN 50000
#define EE 800000
#define E2T (EE + NN)
#define F 128
#define NEG_SLOPE 0.2f
#define BN_EPS 1e-5f

typedef __attribute__((ext_vector_type(2))) float v2f;
typedef __attribute__((ext_vector_type(8))) float v8f;

// Monotonic unsigned encoding of float for atomicMax-based segment-max.
__device__ __forceinline__ unsigned fenc(float x) {
  unsigned u = __float_as_uint(x);
  return (u & 0x80000000u) ? ~u : (u | 0x80000000u);
}
__device__ __forceinline__ float fdec(unsigned u) {
  return (u & 0x80000000u) ? __uint_as_float(u & 0x7fffffffu) : __uint_as_float(~u);
}

__global__ void fill_u32_k(unsigned* __restrict__ p, unsigned v, long n) {
  long i = (long)blockIdx.x * blockDim.x + threadIdx.x;
  if (i < n) p[i] = v;
}

__global__ void sum_all_k(const float* __restrict__ x, float* __restrict__ out, int n) {
  int stride = gridDim.x * blockDim.x;
  float s = 0.f;
  for (int i = blockIdx.x * blockDim.x + threadIdx.x; i < n; i += stride) s += x[i];
#pragma unroll
  for (int m = 1; m < 32; m <<= 1) s += __shfl_xor(s, m, 32);
  if ((threadIdx.x & 31) == 0) atomicAdd(out, s);
}

// O[nrows,128] = Hm[nrows,128] @ W[128,128] using V_WMMA_F32_16X16X4_F32.
// One wave computes one 16x16 output tile; K loop of 32 WMMA steps.
// A 16x4 layout (ISA 7.12.2): lanes 0-15 hold K=0,1 in VGPR0,1; lanes 16-31 hold K=2,3.
// B 4x16: VGPR0 = rows K=0 (lanes 0-15) / K=2 (lanes 16-31), VGPR1 = K=1 / K=3.
// D 16x16: VGPR v = row v (lanes 0-15) / row v+8 (lanes 16-31), N = lane%16.
__global__ void gemm_wmma_f32_k(const float* __restrict__ Hm, const float* __restrict__ W,
                                float* __restrict__ O, int nrows) {
  int wave = (int)(((long)blockIdx.x * blockDim.x + threadIdx.x) >> 5);
  int lane = threadIdx.x & 31;
  int totalTiles = (nrows >> 4) * (F >> 4);
  if (wave >= totalTiles) return;
  int rt = wave >> 3;           // 8 column tiles (F/16)
  int ct = wave & 7;
  int row0 = rt << 4, col0 = ct << 4;
  int half = lane >> 4, lr = lane & 15;
  const float* arow = Hm + (size_t)(row0 + lr) * F + 2 * half;
  const float* bcol = W + (size_t)(2 * half) * F + col0 + lr;
  v8f acc = {};
#pragma unroll
  for (int k0 = 0; k0 < F; k0 += 4) {
    v2f a, b;
    a.x = arow[k0];
    a.y = arow[k0 + 1];
    b.x = bcol[(size_t)k0 * F];
    b.y = bcol[(size_t)(k0 + 1) * F];
    acc = __builtin_amdgcn_wmma_f32_16x16x4_f32(false, a, false, b, (short)0, acc,
                                                false, false);
  }
  float* orow = O + (size_t)(row0 + 8 * half) * F + col0 + lr;
#pragma unroll
  for (int v = 0; v < 8; ++v) orow[(size_t)v * F] = acc[v];
}

// One wave per edge. Lane handles channels 4*lane..4*lane+3 (always within one head
// since 4l..4l+3 never crosses a 32 boundary). Computes GATv2 logit per head,
// stores logits and does segment-max via atomicMax on encoded floats.
__global__ void edge_logits_k(const float* __restrict__ XLm, const float* __restrict__ XRm,
                              const int* __restrict__ src, const int* __restrict__ dst,
                              const float* __restrict__ ea, const float* __restrict__ easum,
                              const float* __restrict__ We, const float* __restrict__ att,
                              float* __restrict__ elog, unsigned* __restrict__ menc,
                              int heads) {
  int wid = (int)(((long)blockIdx.x * blockDim.x + threadIdx.x) >> 5);
  int lane = threadIdx.x & 31;
  if (wid >= E2T) return;
  int s, d;
  float eav;
  if (wid < EE) { s = src[wid]; d = dst[wid]; eav = ea[wid]; }
  else { s = wid - EE; d = s; eav = easum[0] * (1.0f / EE); }  // self loop, mean attr
  int c0 = lane << 2;
  float4 xl = *(const float4*)(XLm + (size_t)s * F + c0);
  float4 xr = *(const float4*)(XRm + (size_t)d * F + c0);
  float4 we = *(const float4*)(We + c0);
  float4 at = *(const float4*)(att + c0);   // att flat index == channel index
  float p = 0.f, v;
  v = xl.x + xr.x + eav * we.x; v = v > 0.f ? v : NEG_SLOPE * v; p += v * at.x;
  v = xl.y + xr.y + eav * we.y; v = v > 0.f ? v : NEG_SLOPE * v; p += v * at.y;
  v = xl.z + xr.z + eav * we.z; v = v > 0.f ? v : NEG_SLOPE * v; p += v * at.z;
  v = xl.w + xr.w + eav * we.w; v = v > 0.f ? v : NEG_SLOPE * v; p += v * at.w;
  // reduce within 8-lane groups (one head each when heads==4)
  p += __shfl_xor(p, 1, 32);
  p += __shfl_xor(p, 2, 32);
  p += __shfl_xor(p, 4, 32);
  if (heads == 1) {
    p += __shfl_xor(p, 8, 32);
    p += __shfl_xor(p, 16, 32);
    if (lane == 0) {
      elog[wid] = p;
      atomicMax(&menc[d], fenc(p));
    }
  } else {
    if ((lane & 7) == 0) {
      int h = lane >> 3;
      elog[(size_t)wid * 4 + h] = p;
      atomicMax(&menc[(size_t)d * 4 + h], fenc(p));
    }
  }
}

// ex = exp(logit - m[dst]); elog <- ex; den[dst,h] += ex
__global__ void edge_expden_k(const int* __restrict__ dst, const unsigned* __restrict__ menc,
                              float* __restrict__ elog, float* __restrict__ den, int heads) {
  long idx = (long)blockIdx.x * blockDim.x + threadIdx.x;
  long tot = (long)E2T * heads;
  if (idx >= tot) return;
  int e = (int)(idx / heads);
  int h = (int)(idx - (long)e * heads);
  int d = (e < EE) ? dst[e] : e - EE;
  float m = fdec(menc[(size_t)d * heads + h]);
  float ex = __expf(elog[idx] - m);
  elog[idx] = ex;
  atomicAdd(&den[(size_t)d * heads + h], ex);
}

// acc[dst] += alpha * xl[src], one wave per edge, float4 per lane, 4 atomics per lane.
__global__ void edge_aggregate_k(const float* __restrict__ XLm, const int* __restrict__ src,
                                 const int* __restrict__ dst, const float* __restrict__ elog,
                                 const float* __restrict__ den, float* __restrict__ acc,
                                 int heads) {
  int wid = (int)(((long)blockIdx.x * blockDim.x + threadIdx.x) >> 5);
  int lane = threadIdx.x & 31;
  if (wid >= E2T) return;
  int s, d;
  if (wid < EE) { s = src[wid]; d = dst[wid]; }
  else { s = wid - EE; d = s; }
  int h = (heads == 4) ? (lane >> 3) : 0;
  float alpha = elog[(size_t)wid * heads + h] /
                (den[(size_t)d * heads + h] + 1e-16f);
  int c0 = lane << 2;
  float4 xl = *(const float4*)(XLm + (size_t)s * F + c0);
  float* a = acc + (size_t)d * F + c0;
  atomicAdd(&a[0], alpha * xl.x);
  atomicAdd(&a[1], alpha * xl.y);
  atomicAdd(&a[2], alpha * xl.z);
  atomicAdd(&a[3], alpha * xl.w);
}

// Per-channel sum / sumsq over rows (128 threads = 128 channels per block).
#define STAT_ROWS 256
__global__ void col_stats_k(const float* __restrict__ acc, float* __restrict__ gsum,
                            float* __restrict__ gsumsq, int nrows) {
  int c = threadIdx.x;
  int r0 = blockIdx.x * STAT_ROWS;
  int r1 = r0 + STAT_ROWS; if (r1 > nrows) r1 = nrows;
  float s = 0.f, ss = 0.f;
  for (int r = r0; r < r1; ++r) {
    float vv = acc[(size_t)r * F + c];
    s += vv; ss += vv * vv;
  }
  atomicAdd(&gsum[c], s);
  atomicAdd(&gsumsq[c], ss);
}

// BatchNorm (training stats, biased var) + ELU. Note: the conv bias cancels inside
// BN ((x+b) - mean(x+b) == x - mean(x), var unchanged), so it is omitted.
__global__ void bn_elu_k(const float* __restrict__ acc, const float* __restrict__ gsum,
                         const float* __restrict__ gsumsq, const float* __restrict__ gamma,
                         const float* __restrict__ beta, float* __restrict__ out, int nrows) {
  long idx = (long)blockIdx.x * blockDim.x + threadIdx.x;
  if (idx >= (long)nrows * F) return;
  int c = (int)(idx & (F - 1));
  float inv_n = 1.0f / nrows;
  float mu = gsum[c] * inv_n;
  float var = gsumsq[c] * inv_n - mu * mu;
  float x = acc[idx];
  float y = (x - mu) * rsqrtf(var + BN_EPS) * gamma[c] + beta[c];
  out[idx] = y > 0.f ? y : (__expf(y) - 1.0f);
}

static inline int cdiv(long a, int b) { return (int)((a + b - 1) / b); }

extern "C" void kernel_launch(void* const* d_in, const int* in_sizes, int n_in,
                              void* d_out, int out_size, void* d_ws, size_t ws_size,
                              hipStream_t stream) {
  (void)in_sizes; (void)n_in; (void)out_size; (void)ws_size;
  const float* x   = (const float*)d_in[0];
  const int*   src = (const int*)d_in[1];
  const int*   dst = (const int*)d_in[2];
  const float* ea  = (const float*)d_in[3];

  // Workspace layout (floats)
  const size_t NF = (size_t)NN * F;
  float*    bufA  = (float*)d_ws;              // layer-1 features
  float*    bufB  = bufA + NF;                 // layer-2 features
  float*    XL    = bufB + NF;                 // source transform
  float*    XR    = XL + NF;                   // target transform
  float*    elog  = XR + NF;                   // [E2T,4] logits -> exp
  unsigned* menc  = (unsigned*)(elog + (size_t)E2T * 4);  // [NN,4] encoded max
  float*    den   = (float*)(menc + (size_t)NN * 4);      // [NN,4]
  float*    gsum  = den + (size_t)NN * 4;      // [128]
  float*    gsq   = gsum + F;                  // [128]
  float*    easum = gsq + F;                   // [1]

  // edge_attr mean (once)
  fill_u32_k<<<1, 32, 0, stream>>>((unsigned*)easum, 0u, 1);
  sum_all_k<<<1024, 256, 0, stream>>>(ea, easum, EE);

  const long perLayerZero = (long)NN * 8 + 2 * F;  // menc + den + gsum + gsq (u32 each)

  for (int i = 0; i < 3; ++i) {
    const int heads = (i == 2) ? 1 : 4;
    const float* hin = (i == 0) ? x : ((i == 1) ? bufA : bufB);
    float* acc = (i == 0) ? bufA : ((i == 1) ? bufB : (float*)d_out);
    const float* Wl    = (const float*)d_in[4 + 7 * i + 0];
    const float* Wr    = (const float*)d_in[4 + 7 * i + 1];
    const float* We    = (const float*)d_in[4 + 7 * i + 2];
    const float* att   = (const float*)d_in[4 + 7 * i + 3];
    const float* gamma = (const float*)d_in[4 + 7 * i + 5];
    const float* beta  = (const float*)d_in[4 + 7 * i + 6];

    // zero segment-max/den/stats and the accumulator
    fill_u32_k<<<cdiv(perLayerZero, 256), 256, 0, stream>>>((unsigned*)menc, 0u, perLayerZero);
    fill_u32_k<<<cdiv((long)NF, 256), 256, 0, stream>>>((unsigned*)acc, 0u, (long)NF);

    // dense transforms via FP32 WMMA
    const long gemmThreads = (long)(NN / 16) * (F / 16) * 32;
    gemm_wmma_f32_k<<<cdiv(gemmThreads, 256), 256, 0, stream>>>(hin, Wl, XL, NN);
    gemm_wmma_f32_k<<<cdiv(gemmThreads, 256), 256, 0, stream>>>(hin, Wr, XR, NN);

    // edge phase
    const long edgeThreads = (long)E2T * 32;
    edge_logits_k<<<cdiv(edgeThreads, 256), 256, 0, stream>>>(
        XL, XR, src, dst, ea, easum, We, att, elog, menc, heads);
    edge_expden_k<<<cdiv((long)E2T * heads, 256), 256, 0, stream>>>(dst, menc, elog, den, heads);
    edge_aggregate_k<<<cdiv(edgeThreads, 256), 256, 0, stream>>>(
        XL, src, dst, elog, den, acc, heads);

    // batchnorm (bias cancels) + ELU, in place
    col_stats_k<<<cdiv((long)NN, STAT_ROWS), F, 0, stream>>>(acc, gsum, gsq, NN);
    bn_elu_k<<<cdiv((long)NN * F, 256), 256, 0, stream>>>(acc, gsum, gsq, gamma, beta, acc, NN);
  }
}